// GMVAE_89567247991444
// MI455X (gfx1250) — compile-verified
//
#include <hip/hip_runtime.h>
#include <hip/hip_bf16.h>
#include <math.h>

// ---------------------------------------------------------------------------
// GMVAE EM prior on MI455X (gfx1250).  Memory-bound: z = 128MB read once per
// EM iteration (both GEMMs fused per tile) => ~640MB total => ~27us roofline
// at 23.3 TB/s.  Both GEMMs use V_WMMA_F32_16X16X4_F32 (full fp32 matrix
// path).  z tiles are double-buffered in LDS via GLOBAL_LOAD_ASYNC_TO_LDS_B128
// (ASYNCcnt) so global latency overlaps the WMMA/softmax phases.
// ---------------------------------------------------------------------------

#define B_       64
#define N_       8192
#define L_       64
#define K_       20
#define K32      32
#define TILE_N   128
#define TILES_PB 4     // tiles per block (double-buffered pipeline)
#define ZSTR     68    // z tile LDS row stride (floats), 16B-aligned rows
#define MSTR     33    // meanT LDS row stride
#define PSTR     131   // postT LDS row stride (odd => conflict-free scatter)

typedef float v2f __attribute__((ext_vector_type(2)));
typedef float v8f __attribute__((ext_vector_type(8)));
typedef int   v4i __attribute__((ext_vector_type(4)));

#if __has_builtin(__builtin_amdgcn_global_load_async_to_lds_b128)
#define HAVE_ASYNC_LDS 1
typedef __attribute__((address_space(1))) v4i* gptr_v4i;
typedef __attribute__((address_space(3))) v4i* lptr_v4i;
#endif

__device__ __forceinline__ v8f wmma_f32(v2f a, v2f b, v8f c) {
  // D = A(16x4) * B(4x16) + C(16x16), all fp32.
  return __builtin_amdgcn_wmma_f32_16x16x4_f32(false, a, false, b, (short)0, c,
                                               false, false);
}

// xor-swizzle within the wave (group-of-32 ds_swizzle: and=0x1f, or=0, xor=M)
template <int XMASK>
__device__ __forceinline__ float swz_xor(float v) {
  return __int_as_float(
      __builtin_amdgcn_ds_swizzle(__float_as_int(v), (XMASK << 10) | 0x1f));
}

// async copy of one 16B chunk per call: global -> LDS (ASYNCcnt tracked)
__device__ __forceinline__ void tile_chunk_to_lds(const float* src, float* dst) {
#if defined(HAVE_ASYNC_LDS)
  __builtin_amdgcn_global_load_async_to_lds_b128(
      (gptr_v4i)(void*)const_cast<float*>(src), (lptr_v4i)dst, 0, 0);
#else
  *reinterpret_cast<float4*>(dst) = *reinterpret_cast<const float4*>(src);
#endif
}

__device__ __forceinline__ void wait_async_lds() {
#if defined(HAVE_ASYNC_LDS)
#if __has_builtin(__builtin_amdgcn_s_wait_asynccnt)
  __builtin_amdgcn_s_wait_asynccnt(0);
#else
  asm volatile("s_wait_asynccnt 0" ::: "memory");
#endif
#endif
}

// issue async load of one 128x64 tile into an LDS buffer (ZSTR row stride)
__device__ __forceinline__ void tile_load(const float* zg_tile, float* lds_buf,
                                          int tid) {
#pragma unroll
  for (int i = 0; i < 8; ++i) {
    int idx = tid + i * 256;  // 2048 float4 total
    int row = idx >> 4, c4 = idx & 15;
    tile_chunk_to_lds(zg_tile + row * L_ + c4 * 4, lds_buf + row * ZSTR + c4 * 4);
  }
}

// ---------------------------------------------------------------------------
// prep: IA = I - A ; M0 = inv(I + IA*IA^T)  (SPD Gauss-Jordan, one block)
// ---------------------------------------------------------------------------
__global__ void prep_kernel(const float* __restrict__ A,
                            float* __restrict__ IAg, float* __restrict__ M0g) {
  __shared__ float ia[64 * 65];
  __shared__ float aug[64 * 130];
  const int t = threadIdx.x;  // 64 threads, thread t owns row t
  for (int j = 0; j < 64; ++j) {
    float v = ((t == j) ? 1.f : 0.f) - A[t * 64 + j];
    ia[t * 65 + j] = v;
    IAg[t * 64 + j] = v;
  }
  __syncthreads();
  for (int j = 0; j < 64; ++j) {
    float s = (t == j) ? 1.f : 0.f;
#pragma unroll 8
    for (int l = 0; l < 64; ++l) s += ia[t * 65 + l] * ia[j * 65 + l];
    aug[t * 130 + j] = s;
    aug[t * 130 + 64 + j] = (t == j) ? 1.f : 0.f;
  }
  __syncthreads();
  for (int p = 0; p < 64; ++p) {
    float invp = 1.f / aug[p * 130 + p];
    __syncthreads();
    aug[p * 130 + t] *= invp;
    aug[p * 130 + 64 + t] *= invp;
    __syncthreads();
    if (t != p) {
      float f = aug[t * 130 + p];
#pragma unroll 8
      for (int c = 0; c < 128; ++c) aug[t * 130 + c] -= f * aug[p * 130 + c];
    }
    __syncthreads();
  }
  for (int j = 0; j < 64; ++j) M0g[t * 64 + j] = aug[t * 130 + 64 + j];
}

// ---------------------------------------------------------------------------
// init pi / trace in d_out
// ---------------------------------------------------------------------------
__global__ void init_out_kernel(float* __restrict__ pi_out,
                                float* __restrict__ trace_out) {
  int i = blockIdx.x * blockDim.x + threadIdx.x;
  if (i < B_ * K_) {
    pi_out[i] = 1.0f / (float)K_;
    trace_out[i] = 1.0f;
  }
}

// ---------------------------------------------------------------------------
// init mean: mean[b,k,:] = z[b, idx[b,k], :] @ M0
// ---------------------------------------------------------------------------
__global__ void init_mean_kernel(const float* __restrict__ z,
                                 const int* __restrict__ init_idx,
                                 const float* __restrict__ M0g,
                                 float* __restrict__ mean_g) {
  __shared__ float raw[64];
  const int k = blockIdx.x, b = blockIdx.y, l = threadIdx.x;
  int idx = init_idx[b * K_ + k];
  raw[l] = z[((size_t)b * N_ + idx) * L_ + l];
  __syncthreads();
  float s = 0.f;
#pragma unroll 8
  for (int m = 0; m < 64; ++m) s += raw[m] * M0g[m * 64 + l];
  mean_g[(b * K_ + k) * L_ + l] = s;
}

// ---------------------------------------------------------------------------
// premean: per (b,k) softmax bias; zero Nk / Msum accumulators
// bias[k] = log pi_k - 0.5*(||mean_k||^2 + ||mean_k @ IA||^2)   (trace==1)
// ---------------------------------------------------------------------------
__global__ void premean_kernel(const float* __restrict__ pi_g,
                               const float* __restrict__ mean_g,
                               const float* __restrict__ IAg,
                               float* __restrict__ biasg,
                               float* __restrict__ Nkg,
                               float* __restrict__ Msum) {
  __shared__ float mrow[64];
  __shared__ float red[64];
  const int k = blockIdx.x, b = blockIdx.y, l = threadIdx.x;
  Msum[((size_t)b * K32 + k) * L_ + l] = 0.f;
  if (k == 0 && l < K32) Nkg[b * K32 + l] = 0.f;
  if (k < K_) {
    mrow[l] = mean_g[(b * K_ + k) * L_ + l];
    __syncthreads();
    float cm = 0.f;
#pragma unroll 8
    for (int m = 0; m < 64; ++m) cm += mrow[m] * IAg[m * 64 + l];
    red[l] = mrow[l] * mrow[l] + cm * cm;
    __syncthreads();
    if (l == 0) {
      float s = 0.f;
      for (int i = 0; i < 64; ++i) s += red[i];
      biasg[b * K32 + k] = __logf(pi_g[b * K_ + k]) - 0.5f * s;
    }
  } else {
    if (l == 0) biasg[b * K32 + k] = -1e30f;
  }
}

// ---------------------------------------------------------------------------
// estep: fused  cross-GEMM -> softmax -> post^T @ z   (WMMA f32 16x16x4)
// grid(16 tile-groups, 64 batches), 256 threads = 8 waves.
// Each block runs TILES_PB tiles with double-buffered async z staging.
// ---------------------------------------------------------------------------
__global__ void __launch_bounds__(256) estep_kernel(
    const float* __restrict__ z, const float* __restrict__ mean_g,
    const float* __restrict__ biasg, float* __restrict__ Nkg,
    float* __restrict__ Msum) {
  __shared__ __align__(16) float z_lds[2][TILE_N * ZSTR];
  __shared__ float meanT[L_ * MSTR];
  __shared__ float postT[K32 * PSTR];
  __shared__ float nk_lds[K32];

  const int tid = threadIdx.x;
  const int b = blockIdx.y;
  const int n_base = blockIdx.x * (TILE_N * TILES_PB);
  const int w = tid >> 5;          // wave 0..7
  const int lane = tid & 31;
  const int half = lane >> 4;      // 0: lanes 0-15, 1: lanes 16-31
  const int j = lane & 15;
  const float* zg_blk = z + ((size_t)b * N_ + n_base) * L_;

  // ---- prologue: kick off tile 0, stage mean^T, zero nk ----
  tile_load(zg_blk, &z_lds[0][0], tid);
  for (int idx = tid; idx < L_ * K32; idx += 256) {
    int l = idx & 63, c = idx >> 6;
    meanT[l * MSTR + c] = (c < K_) ? mean_g[(b * K_ + c) * L_ + l] : 0.f;
  }
  if (tid < K32) nk_lds[tid] = 0.f;
  const float bias0 = biasg[b * K32 + j];
  const float bias1 = biasg[b * K32 + 16 + j];
  wait_async_lds();
  __syncthreads();

  for (int t = 0; t < TILES_PB; ++t) {
    const int cur = t & 1;
    const float* zb = &z_lds[cur][0];
    // prefetch next tile into the other buffer (overlaps with compute below)
    if (t + 1 < TILES_PB)
      tile_load(zg_blk + (size_t)(t + 1) * TILE_N * L_, &z_lds[1 - cur][0], tid);

    // ---- phase 2: cross = z_tile @ meanT, 16-row stripe per wave ----
    v8f acc0 = {0.f, 0.f, 0.f, 0.f, 0.f, 0.f, 0.f, 0.f};
    v8f acc1 = acc0;
#pragma unroll 4
    for (int k0 = 0; k0 < 64; k0 += 4) {
      int krow = k0 + 2 * half;
      const float* ap = &zb[(w * 16 + j) * ZSTR + krow];
      v2f a;  a.x = ap[0];                        a.y = ap[1];
      v2f b0; b0.x = meanT[krow * MSTR + j];      b0.y = meanT[(krow + 1) * MSTR + j];
      v2f b1; b1.x = meanT[krow * MSTR + 16 + j]; b1.y = meanT[(krow + 1) * MSTR + 16 + j];
      acc0 = wmma_f32(a, b0, acc0);
      acc1 = wmma_f32(a, b1, acc1);
    }

    // ---- softmax over K (padded to 32), write post^T, accumulate Nk ----
    float nkp0 = 0.f, nkp1 = 0.f;
#pragma unroll
    for (int r = 0; r < 8; ++r) {
      float l0 = acc0[r] + bias0;
      float l1 = acc1[r] + bias1;
      float mx = fmaxf(l0, l1);
      mx = fmaxf(mx, swz_xor<1>(mx));
      mx = fmaxf(mx, swz_xor<2>(mx));
      mx = fmaxf(mx, swz_xor<4>(mx));
      mx = fmaxf(mx, swz_xor<8>(mx));
      float e0 = __expf(l0 - mx), e1 = __expf(l1 - mx);
      float s = e0 + e1;
      s += swz_xor<1>(s);
      s += swz_xor<2>(s);
      s += swz_xor<4>(s);
      s += swz_xor<8>(s);
      float inv = 1.f / s;
      e0 *= inv; e1 *= inv;
      int nloc = w * 16 + r + 8 * half;   // C layout: rows r / r+8 per half
      postT[j * PSTR + nloc] = e0;        // component j
      postT[(16 + j) * PSTR + nloc] = e1; // component 16+j
      nkp0 += e0; nkp1 += e1;
    }
    nkp0 += swz_xor<16>(nkp0);
    nkp1 += swz_xor<16>(nkp1);
    if (lane < 16) {
      atomicAdd(&nk_lds[j], nkp0);
      atomicAdd(&nk_lds[16 + j], nkp1);
    }
    __syncthreads();
    if (tid < K32) {
      atomicAdd(&Nkg[b * K32 + tid], nk_lds[tid]);
      nk_lds[tid] = 0.f;  // ready for next tile
    }

    // ---- phase 3: Msum += post^T(32x128) @ z_tile(128x64), 16x16 per wave --
    const int Mt = w >> 2, Nt = w & 3;
    v8f acc = {0.f, 0.f, 0.f, 0.f, 0.f, 0.f, 0.f, 0.f};
#pragma unroll 8
    for (int k0 = 0; k0 < TILE_N; k0 += 4) {
      int kk = k0 + 2 * half;
      const float* pp = &postT[(Mt * 16 + j) * PSTR + kk];
      v2f a;  a.x = pp[0];  a.y = pp[1];
      v2f bv; bv.x = zb[kk * ZSTR + Nt * 16 + j];
              bv.y = zb[(kk + 1) * ZSTR + Nt * 16 + j];
      acc = wmma_f32(a, bv, acc);
    }
#pragma unroll
    for (int r = 0; r < 8; ++r) {
      int comp = Mt * 16 + r + 8 * half;
      if (comp < K_)
        atomicAdd(&Msum[((size_t)b * K32 + comp) * L_ + Nt * 16 + j], acc[r]);
    }

    // next tile's data must be resident; postT must be fully consumed
    wait_async_lds();
    __syncthreads();
  }
}

// ---------------------------------------------------------------------------
// mstep: pi = Nk / sum(Nk)   (Nk clamped where zero, matching reference)
// ---------------------------------------------------------------------------
__global__ void mstep_pi_kernel(const float* __restrict__ Nkg,
                                float* __restrict__ pi_g) {
  const int b = blockIdx.x, t = threadIdx.x;  // 32 threads
  float v = (t < K_) ? Nkg[b * K32 + t] : 0.f;
  if (t < K_ && v == 0.f) v = 1e-22f;
  float s = v;
  s += __shfl_xor(s, 1, 32);
  s += __shfl_xor(s, 2, 32);
  s += __shfl_xor(s, 4, 32);
  s += __shfl_xor(s, 8, 32);
  s += __shfl_xor(s, 16, 32);
  if (t < K_) pi_g[b * K_ + t] = v / s;
}

// ---------------------------------------------------------------------------
// mstep: mean = (Msum / Nk) @ M0
// ---------------------------------------------------------------------------
__global__ void mstep_mean_kernel(const float* __restrict__ Nkg,
                                  const float* __restrict__ Msum,
                                  const float* __restrict__ M0g,
                                  float* __restrict__ mean_g) {
  __shared__ float raw[64];
  const int k = blockIdx.x, b = blockIdx.y, l = threadIdx.x;
  float nk = Nkg[b * K32 + k];
  if (nk == 0.f) nk = 1e-22f;
  raw[l] = Msum[((size_t)b * K32 + k) * L_ + l] / nk;
  __syncthreads();
  float s = 0.f;
#pragma unroll 8
  for (int m = 0; m < 64; ++m) s += raw[m] * M0g[m * 64 + l];
  mean_g[(b * K_ + k) * L_ + l] = s;
}

// ---------------------------------------------------------------------------
// launch
// ---------------------------------------------------------------------------
extern "C" void kernel_launch(void* const* d_in, const int* in_sizes, int n_in,
                              void* d_out, int out_size, void* d_ws,
                              size_t ws_size, hipStream_t stream) {
  const float* z = (const float*)d_in[0];
  const float* A = (const float*)d_in[1];
  const int* init_idx = (const int*)d_in[2];

  float* pi_out = (float*)d_out;                 // [64,20]
  float* mean_out = pi_out + B_ * K_;            // [64,20,64]
  float* trace_out = mean_out + B_ * K_ * L_;    // [64,20]

  // workspace (floats): IA 4096 | M0 4096 | bias 2048 | Nk 2048 | Msum 131072
  float* w = (float*)d_ws;
  float* IAg = w;
  float* M0g = IAg + 64 * 64;
  float* biasg = M0g + 64 * 64;
  float* Nkg = biasg + B_ * K32;
  float* Msum = Nkg + B_ * K32;

  prep_kernel<<<1, 64, 0, stream>>>(A, IAg, M0g);
  init_out_kernel<<<(B_ * K_ + 255) / 256, 256, 0, stream>>>(pi_out, trace_out);
  init_mean_kernel<<<dim3(K_, B_), 64, 0, stream>>>(z, init_idx, M0g, mean_out);

  for (int iter = 0; iter < 5; ++iter) {
    premean_kernel<<<dim3(K32, B_), 64, 0, stream>>>(pi_out, mean_out, IAg,
                                                     biasg, Nkg, Msum);
    estep_kernel<<<dim3(N_ / (TILE_N * TILES_PB), B_), 256, 0, stream>>>(
        z, mean_out, biasg, Nkg, Msum);
    mstep_pi_kernel<<<B_, 32, 0, stream>>>(Nkg, pi_out);
    mstep_mean_kernel<<<dim3(K_, B_), 64, 0, stream>>>(Nkg, Msum, M0g,
                                                       mean_out);
  }
}